// DatatoTaskLayer_7095285973616
// MI455X (gfx1250) — compile-verified
//
#include <hip/hip_runtime.h>
#include <hip/hip_bf16.h>
#include <stdint.h>

typedef __attribute__((ext_vector_type(2))) float v2f;
typedef __attribute__((ext_vector_type(8))) float v8f;

#define N_DATA  50000
#define N_TASKS 100000
#define N_EDGES 1000000
#define F_DATA  5
#define F_TASK  12
#define HID     64

// ---- order-preserving float<->uint encoding for atomicMax on floats ----
__device__ __forceinline__ unsigned enc_f32(float x) {
    unsigned u = __float_as_uint(x);
    return (u & 0x80000000u) ? ~u : (u | 0x80000000u);
}
__device__ __forceinline__ float dec_f32(unsigned u) {
    return (u & 0x80000000u) ? __uint_as_float(u & 0x7FFFFFFFu) : __uint_as_float(~u);
}

// ---- fold attention vectors through tiny weight matrices ----
// vsmall[0..11]  = W_dst  @ att_dst   (12)
// vsmall[12..14] = W_edge @ att_edge  (3)
__global__ void k_attvec(const float* __restrict__ Wd, const float* __restrict__ ad,
                         const float* __restrict__ We, const float* __restrict__ ae,
                         float* __restrict__ vsmall) {
    int t = threadIdx.x;
    if (t < 12) {
        float s = 0.f;
        for (int h = 0; h < HID; ++h) s += Wd[t * HID + h] * ad[h];
        vsmall[t] = s;
    } else if (t < 15) {
        int k = t - 12;
        float s = 0.f;
        for (int h = 0; h < HID; ++h) s += We[k * HID + h] * ae[h];
        vsmall[t] = s;
    }
}

// ---- one wave computes a 16xHID tile of X[MxF] @ W[FxHID] via f32 WMMA ----
// A frag (16x4 f32): lanes 0-15 hold K=4s+{0,1}, lanes 16-31 hold K=4s+{2,3}
// B frag (4x16 f32): same K split across lane halves, N striped over 16 lanes
// K-padding uses clamped-address loads + select (no EXEC-divergent loads, so
// the whole K-loop stays branch-free and WMMA sees EXEC all-ones).
template <int F, int KSTEPS>
__device__ __forceinline__ void gemm_tile_wmma(const float* __restrict__ X,
                                               const float* __restrict__ W,
                                               int m0, int lane, v8f c[4]) {
    const int half = lane >> 4;
    const int l15  = lane & 15;
    const int mrow = m0 + l15;
#pragma unroll
    for (int s = 0; s < KSTEPS; ++s) {
        const int  ka  = s * 4 + half * 2;
        const bool v0  = (ka     < F);
        const bool v1  = (ka + 1 < F);
        const int  ka0 = v0 ? ka     : 0;   // clamped, always-valid addresses
        const int  ka1 = v1 ? ka + 1 : 0;
        const float ax = X[mrow * F + ka0];
        const float ay = X[mrow * F + ka1];
        v2f a;
        a.x = v0 ? ax : 0.0f;               // v_cndmask, not exec-masked load
        a.y = v1 ? ay : 0.0f;
#pragma unroll
        for (int t = 0; t < 4; ++t) {
            const int col = t * 16 + l15;
            const float bx = W[ka0 * HID + col];
            const float by = W[ka1 * HID + col];
            v2f b;
            b.x = v0 ? bx : 0.0f;
            b.y = v1 ? by : 0.0f;
            c[t] = __builtin_amdgcn_wmma_f32_16x16x4_f32(
                false, a, false, b, (short)0, c[t], false, false);
        }
    }
}

// ---- h_src = x_data @ W_src ; a_src = h_src @ att_src (fused) ----
__global__ void k_proj_data(const float* __restrict__ x, const float* __restrict__ W,
                            const float* __restrict__ att,
                            float* __restrict__ h, float* __restrict__ arow) {
    const int wave = (blockIdx.x * blockDim.x + threadIdx.x) >> 5;
    const int lane = threadIdx.x & 31;
    if (wave >= N_DATA / 16) return;            // wave-uniform: EXEC stays full
    const int m0 = wave * 16;
    v8f c[4] = {};
    gemm_tile_wmma<F_DATA, 2>(x, W, m0, lane, c);
    const int half = lane >> 4, l15 = lane & 15;
#pragma unroll
    for (int r = 0; r < 8; ++r) {
        const int row = m0 + r + half * 8;      // C layout: VGPR r = M=r (lo half) / M=r+8 (hi half)
        float p = 0.0f;
#pragma unroll
        for (int t = 0; t < 4; ++t) {
            const int col = t * 16 + l15;
            const float v = c[t][r];
            h[row * HID + col] = v;
            p += v * att[col];
        }
        p += __shfl_xor(p, 1);
        p += __shfl_xor(p, 2);
        p += __shfl_xor(p, 4);
        p += __shfl_xor(p, 8);                  // reduce over 16 lanes of each half
        if (l15 == 0) arow[row] = p;
    }
}

// ---- d_out = x_tasks @ W_res + bias (WMMA) ----
__global__ void k_res_proj(const float* __restrict__ xt, const float* __restrict__ Wr,
                           const float* __restrict__ bias, float* __restrict__ out) {
    const int wave = (blockIdx.x * blockDim.x + threadIdx.x) >> 5;
    const int lane = threadIdx.x & 31;
    if (wave >= N_TASKS / 16) return;
    const int m0 = wave * 16;
    v8f c[4] = {};
    gemm_tile_wmma<F_TASK, 3>(xt, Wr, m0, lane, c);
    const int half = lane >> 4, l15 = lane & 15;
#pragma unroll
    for (int r = 0; r < 8; ++r) {
        const int row = m0 + r + half * 8;
#pragma unroll
        for (int t = 0; t < 4; ++t) {
            const int col = t * 16 + l15;
            out[row * HID + col] = c[t][r] + bias[col];
        }
    }
}

// ---- a_dst[t] = x_tasks[t,:] . vsmall[0:12] ----
__global__ void k_adst(const float* __restrict__ xt, const float* __restrict__ vsmall,
                       float* __restrict__ adst) {
    const int t = blockIdx.x * blockDim.x + threadIdx.x;
    if (t >= N_TASKS) return;
    float s = 0.f;
#pragma unroll
    for (int k = 0; k < F_TASK; ++k) s += xt[t * F_TASK + k] * vsmall[k];
    adst[t] = s;
}

__global__ void k_init(unsigned* __restrict__ amax, float* __restrict__ denom) {
    const int t = blockIdx.x * blockDim.x + threadIdx.x;
    if (t < N_TASKS) { amax[t] = 0u; denom[t] = 0.f; }   // enc(-inf) < 0x00800000, so 0 works
}

// ---- edge pass 1: alpha + leaky(0.2), segment max via encoded atomicMax ----
__global__ void k_alpha(const int* __restrict__ src, const int* __restrict__ dst,
                        const float* __restrict__ eattr, const float* __restrict__ asrc,
                        const float* __restrict__ adst, const float* __restrict__ vsmall,
                        float* __restrict__ alpha, unsigned* __restrict__ amax) {
    const int e = blockIdx.x * blockDim.x + threadIdx.x;
    if (e >= N_EDGES) return;
    const int s = src[e], d = dst[e];
    float al = asrc[s] + adst[d]
             + eattr[e * 3 + 0] * vsmall[12]
             + eattr[e * 3 + 1] * vsmall[13]
             + eattr[e * 3 + 2] * vsmall[14];
    al = (al > 0.f) ? al : 0.2f * al;
    alpha[e] = al;
    atomicMax(&amax[d], enc_f32(al));
}

// ---- edge pass 2: ex = exp(alpha - amax[dst]); denom[dst] += ex ----
__global__ void k_expsum(const int* __restrict__ dst, const unsigned* __restrict__ amax,
                         float* __restrict__ alpha, float* __restrict__ denom) {
    const int e = blockIdx.x * blockDim.x + threadIdx.x;
    if (e >= N_EDGES) return;
    const int d = dst[e];
    const float ex = __expf(alpha[e] - dec_f32(amax[d]));
    alpha[e] = ex;                                // reuse buffer: now holds ex
    atomicAdd(&denom[d], ex);
}

// ---- edge pass 3: wave per edge; out[dst] += (ex/denom) * h_src[src] ----
__global__ void k_scatter(const int* __restrict__ src, const int* __restrict__ dst,
                          const float* __restrict__ ex, const float* __restrict__ denom,
                          const float* __restrict__ h, float* __restrict__ out) {
    const int wave = (blockIdx.x * blockDim.x + threadIdx.x) >> 5;
    const int lane = threadIdx.x & 31;
    if (wave >= N_EDGES) return;
    const int s = src[wave], d = dst[wave];
    const float w = ex[wave] / (denom[d] + 1e-16f);
    atomicAdd(&out[d * HID + lane],      w * h[s * HID + lane]);
    atomicAdd(&out[d * HID + lane + 32], w * h[s * HID + lane + 32]);
}

// ---- LayerNorm(64) + leaky(0.01), wave per row, in place ----
__global__ void k_lnorm(float* __restrict__ out, const float* __restrict__ gamma,
                        const float* __restrict__ beta) {
    const int wave = (blockIdx.x * blockDim.x + threadIdx.x) >> 5;
    const int lane = threadIdx.x & 31;
    if (wave >= N_TASKS) return;
    const float x0 = out[wave * HID + lane];
    const float x1 = out[wave * HID + lane + 32];
    float s = x0 + x1;
    float q = x0 * x0 + x1 * x1;
#pragma unroll
    for (int off = 1; off < 32; off <<= 1) {
        s += __shfl_xor(s, off);
        q += __shfl_xor(q, off);
    }
    const float mean = s * (1.0f / 64.0f);
    const float var  = q * (1.0f / 64.0f) - mean * mean;
    const float rstd = rsqrtf(var + 1e-5f);
    float y0 = (x0 - mean) * rstd * gamma[lane]      + beta[lane];
    float y1 = (x1 - mean) * rstd * gamma[lane + 32] + beta[lane + 32];
    y0 = (y0 > 0.f) ? y0 : 0.01f * y0;
    y1 = (y1 > 0.f) ? y1 : 0.01f * y1;
    out[wave * HID + lane]      = y0;
    out[wave * HID + lane + 32] = y1;
}

extern "C" void kernel_launch(void* const* d_in, const int* in_sizes, int n_in,
                              void* d_out, int out_size, void* d_ws, size_t ws_size,
                              hipStream_t stream) {
    const float* x_data    = (const float*)d_in[0];
    const float* x_tasks   = (const float*)d_in[1];
    const int*   edge_idx  = (const int*)d_in[2];
    const float* edge_attr = (const float*)d_in[3];
    const float* W_src     = (const float*)d_in[4];
    const float* W_dst     = (const float*)d_in[5];
    const float* W_edge    = (const float*)d_in[6];
    const float* att_src   = (const float*)d_in[7];
    const float* att_dst   = (const float*)d_in[8];
    const float* att_edge  = (const float*)d_in[9];
    const float* W_res     = (const float*)d_in[10];
    const float* bias      = (const float*)d_in[11];
    const float* ln_gamma  = (const float*)d_in[12];
    const float* ln_beta   = (const float*)d_in[13];
    const int* src = edge_idx;
    const int* dst = edge_idx + N_EDGES;
    float* out = (float*)d_out;

    // workspace layout (floats): ~18.2 MB total
    float*    h_src  = (float*)d_ws;                       // N_DATA*HID
    float*    a_src  = h_src + (size_t)N_DATA * HID;       // N_DATA
    float*    a_dst  = a_src + N_DATA;                     // N_TASKS
    unsigned* amax   = (unsigned*)(a_dst + N_TASKS);       // N_TASKS
    float*    denom  = (float*)(amax + N_TASKS);           // N_TASKS
    float*    alpha  = denom + N_TASKS;                    // N_EDGES (alpha, then ex)
    float*    vsmall = alpha + N_EDGES;                    // 16

    k_attvec<<<1, 64, 0, stream>>>(W_dst, att_dst, W_edge, att_edge, vsmall);

    {   // WMMA projection of data nodes: 3125 tiles, 4 waves/block
        const int waves = N_DATA / 16;
        k_proj_data<<<(waves * 32 + 127) / 128, 128, 0, stream>>>(
            x_data, W_src, att_src, h_src, a_src);
    }
    k_adst<<<(N_TASKS + 255) / 256, 256, 0, stream>>>(x_tasks, vsmall, a_dst);
    k_init<<<(N_TASKS + 255) / 256, 256, 0, stream>>>(amax, denom);
    k_alpha<<<(N_EDGES + 255) / 256, 256, 0, stream>>>(
        src, dst, edge_attr, a_src, a_dst, vsmall, alpha, amax);
    k_expsum<<<(N_EDGES + 255) / 256, 256, 0, stream>>>(dst, amax, alpha, denom);

    {   // WMMA residual projection directly into d_out (initializes every element)
        const int waves = N_TASKS / 16;
        k_res_proj<<<(waves * 32 + 127) / 128, 128, 0, stream>>>(
            x_tasks, W_res, bias, out);
    }
    {   // wave-per-edge weighted scatter into d_out
        const long long thr = (long long)N_EDGES * 32;
        k_scatter<<<(int)((thr + 255) / 256), 256, 0, stream>>>(
            src, dst, alpha, denom, h_src, out);
    }
    k_lnorm<<<(N_TASKS * 32 + 255) / 256, 256, 0, stream>>>(out, ln_gamma, ln_beta);
}